// FastTextSkipgram_43250320671119
// MI455X (gfx1250) — compile-verified
//
#include <hip/hip_runtime.h>
#include <hip/hip_bf16.h>

#define DIM   128
#define NSUB  20
#define NNEG  5
#define BPW   16          // batch elements per workgroup (one wave each)
#define ROWF  132         // padded LDS row stride (floats) -> bank-conflict-free
#define VECS  7           // u, v, 5 negs
#define WGSZ  (BPW * 32)  // 512 threads

typedef float v2f __attribute__((ext_vector_type(2)));
typedef float v8f __attribute__((ext_vector_type(8)));

__device__ __forceinline__ float log_sigmoid(float x) {
    // stable: min(x,0) - log1p(exp(-|x|))
    return fminf(x, 0.0f) - log1pf(expf(-fabsf(x)));
}

__global__ __launch_bounds__(WGSZ)
void fasttext_skipgram_kernel(const int*   __restrict__ u_pos,
                              const int*   __restrict__ v_pos,
                              const int*   __restrict__ v_neg,
                              const float* __restrict__ u_weight,
                              const float* __restrict__ v_weight,
                              float*       __restrict__ per_b_loss)
{
    // LDS: [BPW][VECS][ROWF] staged mean vectors + 256-float WMMA diag scratch
    __shared__ float smem[BPW * VECS * ROWF + 32 * 8];

    const int tid  = threadIdx.x;
    const int wave = tid >> 5;   // 0..15 -> local batch element
    const int lane = tid & 31;   // lane l owns dims 4l..4l+3
    const int b    = blockIdx.x * BPW + wave;

    // ---------------- Phase 1: gather + mean (one wave per b) ----------------
    float aU0=0,aU1=0,aU2=0,aU3=0;
    float aV0=0,aV1=0,aV2=0,aV3=0;
    float aN0[NNEG], aN1[NNEG], aN2[NNEG], aN3[NNEG];
    #pragma unroll
    for (int n = 0; n < NNEG; ++n) { aN0[n]=0; aN1[n]=0; aN2[n]=0; aN3[n]=0; }

    const float4* uw4 = (const float4*)u_weight;
    const float4* vw4 = (const float4*)v_weight;

    #pragma unroll 2
    for (int s = 0; s < NSUB; ++s) {
        const int iu = u_pos[b * NSUB + s];          // wave-uniform -> s_load
        const int iv = v_pos[b * NSUB + s];
        float4 xu = uw4[(size_t)iu * 32 + lane];     // 512B coalesced per wave
        float4 xv = vw4[(size_t)iv * 32 + lane];
        if (iu != 0) {                               // padding_idx=0 on u only
            aU0 += xu.x; aU1 += xu.y; aU2 += xu.z; aU3 += xu.w;
        }
        aV0 += xv.x; aV1 += xv.y; aV2 += xv.z; aV3 += xv.w;
        #pragma unroll
        for (int n = 0; n < NNEG; ++n) {
            const int ig = v_neg[(b * NSUB + s) * NNEG + n];
            float4 xg = vw4[(size_t)ig * 32 + lane];
            aN0[n] += xg.x; aN1[n] += xg.y; aN2[n] += xg.z; aN3[n] += xg.w;
        }
    }

    const float inv = 1.0f / (float)NSUB;
    float* base = &smem[wave * VECS * ROWF];
    const int d = lane * 4;
    { float4 t = { aU0*inv, aU1*inv, aU2*inv, aU3*inv };
      *(float4*)&base[0 * ROWF + d] = t; }
    { float4 t = { aV0*inv, aV1*inv, aV2*inv, aV3*inv };
      *(float4*)&base[1 * ROWF + d] = t; }
    #pragma unroll
    for (int n = 0; n < NNEG; ++n) {
        float4 t = { aN0[n]*inv, aN1[n]*inv, aN2[n]*inv, aN3[n]*inv };
        *(float4*)&base[(2 + n) * ROWF + d] = t;
    }

    __syncthreads();

    // ------------- Phase 2: wave 0 computes all 6 score vectors -------------
    // D(16x16) = Eu(16x128) x Xj^T(128x16), chained v_wmma_f32_16x16x4_f32;
    // we only need diag(D)[t] = dot(embed_u[t], Xj[t]).
    if (tid < 32) {
        const int L    = lane;
        const int m    = L & 15;              // A row / B col supplied by lane
        const int koff = (L >> 4) * 2;        // K sub-offset per ISA layout
        float* scratch = &smem[BPW * VECS * ROWF];
        float scores[6];

        #pragma unroll
        for (int j = 0; j < 6; ++j) {
            v8f c = {0.f,0.f,0.f,0.f,0.f,0.f,0.f,0.f};
            #pragma unroll
            for (int k0 = 0; k0 < DIM; k0 += 4) {
                // A: embed_u rows, B: Xj rows (v=1(v_pos) or neg slot j-1)
                v2f a  = *(const v2f*)&smem[m * (VECS*ROWF) + k0 + koff];
                v2f bb = *(const v2f*)&smem[m * (VECS*ROWF) + (1 + j) * ROWF + k0 + koff];
                c = __builtin_amdgcn_wmma_f32_16x16x4_f32(
                        false, a, false, bb, (short)0, c, false, false);
            }
            // spill D to LDS, pull the diagonal element for thread t = L&15:
            // element (t,t) lives in VGPR t&7 of lane t + ((t>>3)<<4)
            #pragma unroll
            for (int r = 0; r < 8; ++r) scratch[L * 8 + r] = c[r];
            asm volatile("s_wait_dscnt 0x0" ::: "memory");
            const int tt = L & 15;
            const int sl = tt + ((tt >> 3) << 4);
            scores[j] = scratch[sl * 8 + (tt & 7)];
            asm volatile("" ::: "memory");
        }

        if (L < 16) {
            float lp = log_sigmoid(scores[0]);          // positive score
            #pragma unroll
            for (int j = 1; j < 6; ++j)
                lp += log_sigmoid(-scores[j]);          // negatives
            per_b_loss[blockIdx.x * BPW + L] = lp;
        }
    }
}

__global__ __launch_bounds__(1024)
void fasttext_reduce_kernel(const float* __restrict__ per_b,
                            float* __restrict__ out, int n, float invB)
{
    __shared__ float red[1024];
    const int t = threadIdx.x;
    float s = 0.0f;
    for (int i = t; i < n; i += 1024) s += per_b[i];
    red[t] = s;
    __syncthreads();
    #pragma unroll
    for (int off = 512; off > 0; off >>= 1) {
        if (t < off) red[t] += red[t + off];
        __syncthreads();
    }
    if (t == 0) out[0] = -red[0] * invB;
}

extern "C" void kernel_launch(void* const* d_in, const int* in_sizes, int n_in,
                              void* d_out, int out_size, void* d_ws, size_t ws_size,
                              hipStream_t stream)
{
    const int*   u_pos    = (const int*)d_in[0];
    const int*   v_pos    = (const int*)d_in[1];
    const int*   v_neg    = (const int*)d_in[2];
    // d_in[3] = batch_size scalar (shape-derived below)
    const float* u_weight = (const float*)d_in[4];
    const float* v_weight = (const float*)d_in[5];

    const int B = in_sizes[0] / NSUB;          // 4096
    float* per_b = (float*)d_ws;               // B floats of scratch

    fasttext_skipgram_kernel<<<dim3(B / BPW), dim3(WGSZ), 0, stream>>>(
        u_pos, v_pos, v_neg, u_weight, v_weight, per_b);
    fasttext_reduce_kernel<<<dim3(1), dim3(1024), 0, stream>>>(
        per_b, (float*)d_out, B, 1.0f / (float)B);
}